// vel_loss_32641751449777
// MI455X (gfx1250) — compile-verified
//
#include <hip/hip_runtime.h>

// ---------------------------------------------------------------------------
// vel-loss multi-scale reduction for MI455X (gfx1250, wave32)
//   d = pre - tar            [N=500000, F=128] f32
//   loss = ( sum_{s in 1,3,9,27} sum_{k < N/s - 1} || sum_{r in seg_k} d_r ||^2 ) / 4
// Memory-bound: 512 MB traffic -> ~22us floor @ 23.3 TB/s.
// Fast path: tiles with r0+27 <= L27 pass every range guard (L27<=L9<=L3<=L1
// <= nrows), so the inner 27-row loop is branch-free: 54 global_load_b128 +
// FMA accumulation that the scheduler can cluster/pipeline freely.
// ---------------------------------------------------------------------------

typedef float vfloat4 __attribute__((ext_vector_type(4)));
typedef float v2f     __attribute__((ext_vector_type(2)));
typedef float v8f     __attribute__((ext_vector_type(8)));

#define FEAT            128
#define FEAT4           32          // 128 floats = 32 float4 per row
#define TILE_ROWS       27          // lcm-aligned tile: whole segments of every step
#define WAVES_PER_BLOCK 8           // 256 threads = 8 waves on wave32

__device__ __forceinline__ vfloat4 vzero4() {
    vfloat4 v; v.x = 0.f; v.y = 0.f; v.z = 0.f; v.w = 0.f; return v;
}
__device__ __forceinline__ float dot4(vfloat4 a) {
    return a.x * a.x + a.y * a.y + a.z * a.z + a.w * a.w;
}

// FULL=true: tile known fully in range for every step -> all guards fold away.
template <bool FULL>
__device__ __forceinline__ float tile_sum(const vfloat4* __restrict__ p4,
                                          const vfloat4* __restrict__ t4,
                                          int r0, int lane,
                                          int nrows, int L1, int L3, int L9, int L27)
{
    vfloat4 a3  = vzero4();
    vfloat4 a9  = vzero4();
    vfloat4 a27 = vzero4();
    float   s   = 0.f;

#pragma unroll
    for (int j = 0; j < TILE_ROWS; ++j) {
        const int r = r0 + j;
        if (FULL || r < nrows) {               // wave-uniform tail guard
            const int idx = r * FEAT4 + lane;  // max ~16M, fits int
            vfloat4 pv = __builtin_nontemporal_load(&p4[idx]);
            vfloat4 tv = __builtin_nontemporal_load(&t4[idx]);
            vfloat4 d;
            d.x = pv.x - tv.x; d.y = pv.y - tv.y;
            d.z = pv.z - tv.z; d.w = pv.w - tv.w;
            if (FULL || r < L1) s += dot4(d);  // step 1: per-row norm
            a3.x += d.x; a3.y += d.y; a3.z += d.z; a3.w += d.w;
            a9.x += d.x; a9.y += d.y; a9.z += d.z; a9.w += d.w;
            a27.x += d.x; a27.y += d.y; a27.z += d.z; a27.w += d.w;
        }
        if ((j % 3) == 2) {                    // step-3 segment closes
            if (FULL || r < L3) s += dot4(a3);
            a3 = vzero4();
        }
        if ((j % 9) == 8) {                    // step-9 segment closes
            if (FULL || r < L9) s += dot4(a9);
            a9 = vzero4();
        }
        if (j == TILE_ROWS - 1) {              // step-27 segment closes
            if (FULL || r < L27) s += dot4(a27);
            a27 = vzero4();
        }
    }
    return s;
}

__global__ __launch_bounds__(256)
void vel_partial_kernel(const float* __restrict__ pre,
                        const float* __restrict__ tar,
                        float* __restrict__ partial,
                        int nrows, int L1, int L3, int L9, int L27)
{
    const int wave = threadIdx.x >> 5;
    const int lane = threadIdx.x & 31;
    const int tile = blockIdx.x * WAVES_PER_BLOCK + wave;
    const int r0   = tile * TILE_ROWS;

    const vfloat4* __restrict__ p4 = (const vfloat4*)pre;
    const vfloat4* __restrict__ t4 = (const vfloat4*)tar;

    float s;
    if (r0 + TILE_ROWS <= L27) {   // L27 is the tightest limit -> all guards pass
        s = tile_sum<true >(p4, t4, r0, lane, nrows, L1, L3, L9, L27);
    } else {
        s = tile_sum<false>(p4, t4, r0, lane, nrows, L1, L3, L9, L27);
    }

    // wave32 butterfly reduction
#pragma unroll
    for (int off = 16; off > 0; off >>= 1)
        s += __shfl_xor(s, off, 32);

    __shared__ float wsum[WAVES_PER_BLOCK];
    if (lane == 0) wsum[wave] = s;
    __syncthreads();
    if (threadIdx.x == 0) {
        float t = 0.f;
#pragma unroll
        for (int i = 0; i < WAVES_PER_BLOCK; ++i) t += wsum[i];
        partial[blockIdx.x] = t;
    }
}

// Final deterministic reduction of the per-block partials on the matrix unit:
// D = ones(16x4) * B(4x16) + C sums all 64 B-elements into each D row,
// independent of which K/N slot each value occupies. Summing c[0] across the
// full wave double-counts (rows M=0 and M=8 identical) -> total = 0.5*sum.
__global__ __launch_bounds__(32)
void vel_final_kernel(const float* __restrict__ partial, int n,
                      float* __restrict__ out)
{
    const int lane = threadIdx.x;   // single wave, EXEC all ones for WMMA

    v2f ones; ones.x = 1.f; ones.y = 1.f;   // A = 16x4 all-ones (2 VGPRs)
    v8f c = {};                              // 16x16 f32 accumulator

    const int iters = (n + 63) >> 6;        // 64 partials per WMMA
    for (int it = 0; it < iters; ++it) {
        const int base = (it << 6) + (lane << 1);
        // clamp index -> unconditional load + v_cndmask (no exec branching)
        const int i0 = (base     < n) ? base     : 0;
        const int i1 = (base + 1 < n) ? base + 1 : 0;
        float v0 = partial[i0];
        float v1 = partial[i1];
        v2f b;
        b.x = (base     < n) ? v0 : 0.f;
        b.y = (base + 1 < n) ? v1 : 0.f;
        c = __builtin_amdgcn_wmma_f32_16x16x4_f32(
                /*neg_a=*/false, ones, /*neg_b=*/false, b,
                /*c_mod=*/(short)0, c, /*reuse_a=*/false, /*reuse_b=*/false);
    }

    float t = c[0];
#pragma unroll
    for (int off = 16; off > 0; off >>= 1)
        t += __shfl_xor(t, off, 32);

    // 0.5 (double-counted wave sum) * 0.25 (reference divides by 4)
    if (lane == 0) out[0] = t * 0.125f;
}

extern "C" void kernel_launch(void* const* d_in, const int* in_sizes, int n_in,
                              void* d_out, int out_size, void* d_ws, size_t ws_size,
                              hipStream_t stream)
{
    const float* pre = (const float*)d_in[0];
    const float* tar = (const float*)d_in[1];
    float* out       = (float*)d_out;
    float* partial   = (float*)d_ws;   // blocks * 4 bytes (~9.3 KB)

    const int nrows = in_sizes[0] / FEAT;          // 500000
    const int L1  = (nrows / 1  - 1) * 1;          // valid-row limits per step
    const int L3  = (nrows / 3  - 1) * 3;
    const int L9  = (nrows / 9  - 1) * 9;
    const int L27 = (nrows / 27 - 1) * 27;

    const int tiles  = (nrows + TILE_ROWS - 1) / TILE_ROWS;
    const int blocks = (tiles + WAVES_PER_BLOCK - 1) / WAVES_PER_BLOCK;

    vel_partial_kernel<<<blocks, 256, 0, stream>>>(pre, tar, partial,
                                                   nrows, L1, L3, L9, L27);
    vel_final_kernel<<<1, 32, 0, stream>>>(partial, blocks, out);
}